// GCN_24764781429371
// MI455X (gfx1250) — compile-verified
//
#include <hip/hip_runtime.h>
#include <hip/hip_bf16.h>

typedef __attribute__((ext_vector_type(2))) float v2f;
typedef __attribute__((ext_vector_type(8))) float v8f;

#define N_NODES  50000
#define N_EDGES  800000
#define IN_SIZE  512
#define HID_SIZE 128
#define OUT_SIZE 64

// ---------------------------------------------------------------- utilities

__global__ void gcn_zero_kernel(float* __restrict__ p, long n) {
    long i = (long)blockIdx.x * blockDim.x + threadIdx.x;
    if (i < n) p[i] = 0.0f;
}

// Accumulate out/in degrees with f32 atomics (L2 atomics, no return).
__global__ void gcn_deg_kernel(const int* __restrict__ src,
                               const int* __restrict__ dst,
                               float* __restrict__ outdeg,
                               float* __restrict__ indeg) {
    int e = blockIdx.x * blockDim.x + threadIdx.x;
    if (e < N_EDGES) {
        atomicAdd(&outdeg[src[e]], 1.0f);
        atomicAdd(&indeg[dst[e]], 1.0f);
    }
}

// norm = rsqrt(max(deg, 1)) in place (DGL clamps degree to >= 1).
__global__ void gcn_norm_kernel(float* __restrict__ outn, float* __restrict__ inn) {
    int i = blockIdx.x * blockDim.x + threadIdx.x;
    if (i < N_NODES) {
        float a = outn[i]; a = a < 1.0f ? 1.0f : a; outn[i] = rsqrtf(a);
        float b = inn[i];  b = b < 1.0f ? 1.0f : b; inn[i]  = rsqrtf(b);
    }
}

// ---------------------------------------------------------------- WMMA GEMM
// Y[M,N] = (A[M,K] * scale[M]) @ W[K,N], fp32 via V_WMMA_F32_16X16X4_F32.
// One wave computes one 16x16 C tile; block = (N/16) waves covering one
// 16-row strip. M must be a multiple of 16 (50000 = 3125*16), K of 4, N of 16.
//
// ISA VGPR layouts (CDNA5 ISA):
//   A 16x4 f32 : lane L -> M = L&15 ; VGPR0 = K(2*(L>>4)), VGPR1 = K(2*(L>>4)+1)
//   B 4x16 f32 : lane L -> N = L&15 ; VGPR0 = K(2*(L>>4)), VGPR1 = K(2*(L>>4)+1)
//   C/D 16x16  : lane L -> N = L&15 ; VGPR r = M(r + 8*(L>>4))
template <int K, int N>
__global__ __launch_bounds__(256) void gcn_gemm_wmma(const float* __restrict__ A,
                                                     const float* __restrict__ W,
                                                     const float* __restrict__ scale,
                                                     float* __restrict__ Y) {
    const int m0   = blockIdx.x * 16;
    const int wave = threadIdx.x >> 5;
    const int lane = threadIdx.x & 31;
    const int n0   = wave * 16;
    const int row  = lane & 15;     // M for A-operand, N for B/C/D operands
    const int hi   = lane >> 4;

    const float s = scale[m0 + row];
    const float* __restrict__ arow = A + (long)(m0 + row) * K;

    v8f c = {};
    #pragma unroll 4
    for (int k = 0; k < K; k += 4) {
        v2f a, b;
        const int ka = k + 2 * hi;
        a.x = arow[ka]     * s;
        a.y = arow[ka + 1] * s;
        b.x = W[(long)ka       * N + n0 + row];
        b.y = W[(long)(ka + 1) * N + n0 + row];
        // 8 args: (neg_a, A, neg_b, B, c_mod, C, reuse_a, reuse_b)
        c = __builtin_amdgcn_wmma_f32_16x16x4_f32(false, a, false, b,
                                                  (short)0, c, false, false);
    }

    #pragma unroll
    for (int r = 0; r < 8; ++r)
        Y[(long)(m0 + r + 8 * hi) * N + n0 + row] = c[r];
}

// ---------------------------------------------------------------- edge scatter
// agg[dst] += Y[src]; one wave per edge, each lane owns D/32 contiguous floats.
// Y rows are L2-resident (25.6 MB / 12.8 MB << 192 MB L2).
template <int D>
__global__ __launch_bounds__(256) void gcn_scatter_kernel(const int* __restrict__ src,
                                                          const int* __restrict__ dst,
                                                          const float* __restrict__ Y,
                                                          float* __restrict__ agg) {
    constexpr int PER = D / 32;                 // floats per lane (4 or 2)
    const int wave = threadIdx.x >> 5;
    const int lane = threadIdx.x & 31;
    const long e = (long)blockIdx.x * 8 + wave; // 256 threads = 8 edges/block
    if (e >= N_EDGES) return;

    const int s = src[e];
    const int d = dst[e];
    const float* __restrict__ ys = Y + (long)s * D + lane * PER;
    float* __restrict__ ad       = agg + (long)d * D + lane * PER;

    float v[PER];
    #pragma unroll
    for (int j = 0; j < PER; ++j) v[j] = ys[j];   // merges into b128/b64 load
    #pragma unroll
    for (int j = 0; j < PER; ++j) atomicAdd(&ad[j], v[j]);
}

// ---------------------------------------------------------------- finalize
// out = act(agg * in_norm[row] + bias[col])
template <int D, bool RELU>
__global__ void gcn_finalize_kernel(const float* __restrict__ agg,
                                    const float* __restrict__ inn,
                                    const float* __restrict__ bias,
                                    float* __restrict__ out) {
    long i = (long)blockIdx.x * blockDim.x + threadIdx.x;
    if (i >= (long)N_NODES * D) return;
    const int r = (int)(i / D);
    const int c = (int)(i % D);
    float v = agg[i] * inn[r] + bias[c];
    if (RELU) v = v > 0.0f ? v : 0.0f;
    out[i] = v;
}

// ---------------------------------------------------------------- launcher

extern "C" void kernel_launch(void* const* d_in, const int* in_sizes, int n_in,
                              void* d_out, int out_size, void* d_ws, size_t ws_size,
                              hipStream_t stream) {
    const float* feat = (const float*)d_in[0];  // [50000, 512]
    const float* W1   = (const float*)d_in[1];  // [512, 128]
    const float* b1   = (const float*)d_in[2];  // [128]
    const float* W2   = (const float*)d_in[3];  // [128, 64]
    const float* b2   = (const float*)d_in[4];  // [64]
    const int*   src  = (const int*)d_in[5];    // [800000]
    const int*   dst  = (const int*)d_in[6];    // [800000]

    float* ws   = (float*)d_ws;
    float* outn = ws;                                  // 50000
    float* inn  = ws + N_NODES;                        // 50000
    float* Y1   = ws + 2 * N_NODES;                    // 50000*128
    float* H1   = Y1 + (long)N_NODES * HID_SIZE;       // 50000*128 (agg1 -> h1 in place)
    float* Y2   = H1 + (long)N_NODES * HID_SIZE;       // 50000*64
    float* AG2  = Y2 + (long)N_NODES * OUT_SIZE;       // 50000*64

    const long nDeg  = 2L * N_NODES;
    const long nAgg1 = (long)N_NODES * HID_SIZE;
    const long nAgg2 = (long)N_NODES * OUT_SIZE;

    // zero accumulators (deterministic every call; graph-capture safe)
    gcn_zero_kernel<<<(int)((nDeg  + 255) / 256), 256, 0, stream>>>(outn, nDeg);
    gcn_zero_kernel<<<(int)((nAgg1 + 255) / 256), 256, 0, stream>>>(H1, nAgg1);
    gcn_zero_kernel<<<(int)((nAgg2 + 255) / 256), 256, 0, stream>>>(AG2, nAgg2);

    // degrees -> norms
    gcn_deg_kernel<<<(N_EDGES + 255) / 256, 256, 0, stream>>>(src, dst, outn, inn);
    gcn_norm_kernel<<<(N_NODES + 255) / 256, 256, 0, stream>>>(outn, inn);

    // layer 1: Y1 = (X * outn) @ W1 ; scatter ; h1 = relu(agg * inn + b1)
    gcn_gemm_wmma<IN_SIZE, HID_SIZE>
        <<<N_NODES / 16, (HID_SIZE / 16) * 32, 0, stream>>>(feat, W1, outn, Y1);
    gcn_scatter_kernel<HID_SIZE>
        <<<(N_EDGES + 7) / 8, 256, 0, stream>>>(src, dst, Y1, H1);
    gcn_finalize_kernel<HID_SIZE, true>
        <<<(int)((nAgg1 + 255) / 256), 256, 0, stream>>>(H1, inn, b1, H1);

    // layer 2: Y2 = (h1 * outn) @ W2 ; scatter ; out = agg * inn + b2
    gcn_gemm_wmma<HID_SIZE, OUT_SIZE>
        <<<N_NODES / 16, (OUT_SIZE / 16) * 32, 0, stream>>>(H1, W2, outn, Y2);
    gcn_scatter_kernel<OUT_SIZE>
        <<<(N_EDGES + 7) / 8, 256, 0, stream>>>(src, dst, Y2, AG2);
    gcn_finalize_kernel<OUT_SIZE, false>
        <<<(int)((nAgg2 + 255) / 256), 256, 0, stream>>>(AG2, inn, b2, (float*)d_out);
}